// ResidualVectorQuantizer_60438779789624
// MI455X (gfx1250) — compile-verified
//
#include <hip/hip_runtime.h>

// ---------------------------------------------------------------------------
// Residual Vector Quantizer for MI455X (gfx1250, wave32, WMMA).
//
// Math per stage:  argmin_k ||res - cb_k||^2  ==  argmin_k (||cb_k||^2 - 2 res.cb_k)
// res.cb computed with exact-fp32 V_WMMA_F32_16X16X4_F32 (A = 16 tokens x 4 dims,
// B = 4 dims x 16 codewords), accumulating over E=64 in 16 chained WMMAs.
// Codebook staged through LDS (128 codewords * pitch-68 floats, bank-conflict-free).
// quant = x - res_final, so only the residual is carried in registers.
// ---------------------------------------------------------------------------

typedef __attribute__((ext_vector_type(2))) float v2f;
typedef __attribute__((ext_vector_type(8))) float v8f;

#define B_   16
#define E_   64
#define HW_  4096          // H*W
#define N_   65536         // B*H*W tokens
#define K_   1024
#define NC_  8

#define WAVES           4
#define TPB             (WAVES * 32)
#define TILES_PER_WAVE  2                       // 2 M-tiles of 16 tokens per wave
#define TOK_PER_BLOCK   (WAVES * TILES_PER_WAVE * 16)   // 128
#define CHUNK           128                     // codewords resident in LDS
#define PITCH           68                      // padded LDS row pitch (floats)

// --- tiny precompute: ||cb_k||^2 for all stages/codewords -> d_ws -------------
__global__ void rvq_norms_kernel(const float* __restrict__ cb,
                                 float* __restrict__ norms) {
    int i = blockIdx.x * blockDim.x + threadIdx.x;
    if (i < NC_ * K_) {
        const float* r = cb + (size_t)i * E_;
        float s = 0.f;
        #pragma unroll
        for (int e = 0; e < E_; ++e) s += r[e] * r[e];
        norms[i] = s;
    }
}

__launch_bounds__(TPB)
__global__ void rvq_main_kernel(const float* __restrict__ emb,
                                const float* __restrict__ cb,
                                const float* __restrict__ cbnorm,
                                float* __restrict__ out) {
    __shared__ float lds_cb[CHUNK * PITCH];                 // ~34.8 KB
    __shared__ int   lds_idx[WAVES][TILES_PER_WAVE][16];

    const int tid  = threadIdx.x;
    const int lane = tid & 31;
    const int wave = tid >> 5;
    const int nl   = lane & 15;          // token-in-tile (A) / codeword-in-tile (B,D)
    const int ke   = (lane >> 4) << 1;   // K-subgroup per half-wave: 0 or 2

    // Per-lane residual in WMMA A-operand layout:
    // res[t][2*ks+c] = x[token(t, nl)][e = 4*ks + ke + c]
    int   base[TILES_PER_WAVE];          // element offset of (b, e=0, hw)
    float res[TILES_PER_WAVE][32];

    #pragma unroll
    for (int t = 0; t < TILES_PER_WAVE; ++t) {
        int tk = blockIdx.x * TOK_PER_BLOCK + wave * (TILES_PER_WAVE * 16) + t * 16 + nl;
        int b  = tk >> 12;               // / HW_
        int hw = tk & 4095;
        base[t] = b * (E_ * HW_) + hw;
        #pragma unroll
        for (int j = 0; j < 16; ++j) {
            int e = 4 * j + ke;
            res[t][2 * j + 0] = emb[base[t] + (e + 0) * HW_];
            res[t][2 * j + 1] = emb[base[t] + (e + 1) * HW_];
        }
    }

    for (int s = 0; s < NC_; ++s) {
        const float* cbs = cb + (size_t)s * K_ * E_;

        float minv[TILES_PER_WAVE][8];
        int   mini[TILES_PER_WAVE][8];
        #pragma unroll
        for (int t = 0; t < TILES_PER_WAVE; ++t)
            #pragma unroll
            for (int r = 0; r < 8; ++r) { minv[t][r] = 3.4e38f; mini[t][r] = 0; }

        for (int c0 = 0; c0 < K_; c0 += CHUNK) {
            __syncthreads();                       // protect lds_cb from prior readers
            for (int i = tid; i < CHUNK * E_; i += TPB) {
                int row = i >> 6, col = i & 63;
                lds_cb[row * PITCH + col] = cbs[(size_t)(c0 + row) * E_ + col];
            }
            __syncthreads();

            for (int nt = 0; nt < CHUNK / 16; ++nt) {
                v8f acc[TILES_PER_WAVE] = {};      // D/C: 8 VGPRs = 16x16 f32 tile
                const float* lrow = &lds_cb[(nt * 16 + nl) * PITCH + ke];

                #pragma unroll
                for (int ks = 0; ks < 16; ++ks) {  // E = 64 in steps of K=4
                    v2f bmat;                      // B 4x16: K=ke..ke+1 per half-wave
                    bmat.x = lrow[4 * ks + 0];
                    bmat.y = lrow[4 * ks + 1];
                    #pragma unroll
                    for (int t = 0; t < TILES_PER_WAVE; ++t) {
                        v2f amat;                  // A 16x4 straight from residual regs
                        amat.x = res[t][2 * ks + 0];
                        amat.y = res[t][2 * ks + 1];
                        acc[t] = __builtin_amdgcn_wmma_f32_16x16x4_f32(
                            false, amat, false, bmat, (short)0, acc[t], false, false);
                    }
                }

                int   ng = c0 + nt * 16 + nl;      // this lane's codeword index
                float cn = cbnorm[s * K_ + ng];
                #pragma unroll
                for (int t = 0; t < TILES_PER_WAVE; ++t)
                    #pragma unroll
                    for (int r = 0; r < 8; ++r) {  // rows m = r (lo half) / r+8 (hi half)
                        float d = cn - 2.0f * acc[t][r];
                        if (d < minv[t][r]) { minv[t][r] = d; mini[t][r] = ng; }
                    }
            }
        }

        // Cross-lane argmin inside each 16-lane half (same token row set).
        #pragma unroll
        for (int t = 0; t < TILES_PER_WAVE; ++t)
            #pragma unroll
            for (int r = 0; r < 8; ++r) {
                float v  = minv[t][r];
                int   ix = mini[t][r];
                #pragma unroll
                for (int m = 1; m <= 8; m <<= 1) {
                    float ov = __shfl_xor(v, m, 32);
                    int   oi = __shfl_xor(ix, m, 32);
                    if (ov < v || (ov == v && oi < ix)) { v = ov; ix = oi; }
                }
                if (lane == 0)  lds_idx[wave][t][r]     = ix;   // tokens m = 0..7
                if (lane == 16) lds_idx[wave][t][r + 8] = ix;   // tokens m = 8..15
            }
        __syncthreads();

        // Residual update: res -= cb[idx]; gather 8B slices from L2-resident codebook.
        #pragma unroll
        for (int t = 0; t < TILES_PER_WAVE; ++t) {
            int ix = lds_idx[wave][t][nl];
            const float* crow = cbs + (size_t)ix * E_ + ke;
            #pragma unroll
            for (int j = 0; j < 16; ++j) {
                float2 g = *(const float2*)(crow + 4 * j);
                res[t][2 * j + 0] -= g.x;
                res[t][2 * j + 1] -= g.y;
            }
        }
    }

    // quant = sum_s q_s = x - res_final
    #pragma unroll
    for (int t = 0; t < TILES_PER_WAVE; ++t)
        #pragma unroll
        for (int j = 0; j < 16; ++j) {
            int e  = 4 * j + ke;
            int a0 = base[t] + (e + 0) * HW_;
            int a1 = base[t] + (e + 1) * HW_;
            out[a0] = emb[a0] - res[t][2 * j + 0];
            out[a1] = emb[a1] - res[t][2 * j + 1];
        }
}

extern "C" void kernel_launch(void* const* d_in, const int* in_sizes, int n_in,
                              void* d_out, int out_size, void* d_ws, size_t ws_size,
                              hipStream_t stream) {
    const float* emb = (const float*)d_in[0];   // [B,E,H,W] fp32
    const float* cb  = (const float*)d_in[1];   // [NC,K,E] fp32
    float* out   = (float*)d_out;               // [B,E,H,W] fp32
    float* norms = (float*)d_ws;                // NC*K fp32 (32 KB)

    rvq_norms_kernel<<<(NC_ * K_ + 255) / 256, 256, 0, stream>>>(cb, norms);
    rvq_main_kernel<<<N_ / TOK_PER_BLOCK, TPB, 0, stream>>>(emb, cb, norms, out);
}